// SoftclDiceAdapter_86294482912077
// MI455X (gfx1250) — compile-verified
//
#include <hip/hip_runtime.h>
#include <hip/hip_bf16.h>
#include <stdint.h>

// ---------------------------------------------------------------------------
// soft-clDice on (2,1,160,160,160). Only channel 1 of the skeletons reaches
// the loss, so we skeletonize p = sigmoid(logits) and tgt = float(target).
//   soft_erode  = 7-point star min  (+inf SAME pad == coordinate clamping)
//   soft_dilate = 27-point max      (-inf SAME pad == ignore OOB)
//   soft_open   = dilate(erode(.)), and the erode chain is shared with it.
// Per field, 4x fused kernels each do: async-DMA a halo-2 tile of `cur` into
// LDS (CDNA5 global_load_async_to_lds_b32 / ASYNCcnt / s_wait_asynccnt),
// compute the halo-1 erode tile in LDS, then the 27-point max + skeleton
// update, writing e (next iteration's cur) as a by-product.
// ---------------------------------------------------------------------------

#define DIM 160
#define VOL (DIM * DIM * DIM)           // 4,096,000
#define NTOT (2 * VOL)                  // 8,192,000
#define TX 32
#define TY 8
#define TZ 8
// erode tile (halo 1 around outputs)
#define EX (TX + 2)                     // 34
#define EY (TY + 2)                     // 10
#define EZ (TZ + 2)                     // 10
#define ESZ (EX * EY * EZ)              // 3400 floats
// cur tile (halo 2 around outputs)
#define CX (TX + 4)                     // 36
#define CY (TY + 4)                     // 12
#define CZ (TZ + 4)                     // 12
#define CSZ (CX * CY * CZ)              // 5184 floats
#define RED_BLOCKS 1024

__device__ __forceinline__ float sigmoidf_dev(float x) {
    return 1.0f / (1.0f + expf(-x));
}

__device__ __forceinline__ int clampi(int v, int lo, int hi) {
    return v < lo ? lo : (v > hi ? hi : v);
}

// CDNA5 async DMA: copy one dword from global memory into LDS (per lane).
// VDST operand = per-lane LDS byte address, VADDR = 64-bit global address.
__device__ __forceinline__ void async_load_dword_to_lds(uint32_t lds_byte_addr,
                                                        const float* gaddr) {
    asm volatile("global_load_async_to_lds_b32 %0, %1, off"
                 :
                 : "v"(lds_byte_addr), "v"(gaddr)
                 : "memory");
}

__device__ __forceinline__ void wait_asynccnt_zero() {
    asm volatile("s_wait_asynccnt 0" ::: "memory");
}

// ---------------------------------------------------------------------------
// dst[i] = mode==0 ? sigmoid(float_in[i]) : float(int_in[i])   (float4 wide)
__global__ __launch_bounds__(256) void init_field_kernel(const void* __restrict__ in,
                                                         float* __restrict__ dst,
                                                         int mode) {
    int i = blockIdx.x * 256 + threadIdx.x;   // float4 index
    if (i >= NTOT / 4) return;
    float4 v;
    if (mode == 0) {
        float4 x = ((const float4*)in)[i];
        v.x = sigmoidf_dev(x.x);
        v.y = sigmoidf_dev(x.y);
        v.z = sigmoidf_dev(x.z);
        v.w = sigmoidf_dev(x.w);
    } else {
        int4 x = ((const int4*)in)[i];
        v.x = (float)x.x;
        v.y = (float)x.y;
        v.z = (float)x.z;
        v.w = (float)x.w;
    }
    ((float4*)dst)[i] = v;
}

// ---------------------------------------------------------------------------
// Fused: e = erode(cur); d = dilate(e); delta = relu(cur - d);
//        skel = first ? delta : skel + relu(delta - skel*delta);
//        if (write_e) store interior of e for the next iteration.
__global__ __launch_bounds__(256) void fused_open_update_kernel(
    const float* __restrict__ cur, float* __restrict__ e_out,
    float* __restrict__ skel, int first, int write_e) {
    __shared__ float ct[CSZ];   // cur tile, halo 2 (clamped coords)
    __shared__ float et[ESZ];   // erode tile, halo 1 (-inf outside volume)

    int b = blockIdx.z / (DIM / TZ);
    int bz = blockIdx.z % (DIM / TZ);
    int x0 = blockIdx.x * TX, y0 = blockIdx.y * TY, z0 = bz * TZ;
    const float* cbase = cur + (size_t)b * VOL;
    int tid = threadIdx.x;

    // Stage halo-2 cur tile via async global->LDS DMA. Clamped coordinates
    // keep every lane's address valid and reproduce +inf min-pool padding.
    for (int h = tid; h < CSZ; h += 256) {
        int hx = h % CX;
        int tt = h / CX;
        int hy = tt % CY;
        int hz = tt / CY;
        int gx = clampi(x0 - 2 + hx, 0, DIM - 1);
        int gy = clampi(y0 - 2 + hy, 0, DIM - 1);
        int gz = clampi(z0 - 2 + hz, 0, DIM - 1);
        const float* gp = cbase + ((size_t)gz * DIM + gy) * DIM + gx;
        async_load_dword_to_lds((uint32_t)(uintptr_t)(&ct[h]), gp);
    }
    wait_asynccnt_zero();   // drain this wave's ASYNCcnt (LDS writes landed)
    __syncthreads();        // tile visible to all 8 waves

    // Erode tile (halo 1). Cells whose global position is outside the volume
    // must be -inf (dilate's padding); a clamped phantom erode would be wrong.
    for (int h = tid; h < ESZ; h += 256) {
        int ex = h % EX;
        int tt = h / EX;
        int ey = tt % EY;
        int ez = tt / EY;
        int gx = x0 - 1 + ex, gy = y0 - 1 + ey, gz = z0 - 1 + ez;
        float v;
        if (gx < 0 || gx >= DIM || gy < 0 || gy >= DIM || gz < 0 || gz >= DIM) {
            v = -INFINITY;
        } else {
            int cx = ex + 1, cy = ey + 1, cz = ez + 1;   // ct coords
            float m = ct[(cz * CY + cy) * CX + cx];
            m = fminf(m, ct[(cz * CY + cy) * CX + (cx - 1)]);
            m = fminf(m, ct[(cz * CY + cy) * CX + (cx + 1)]);
            m = fminf(m, ct[(cz * CY + (cy - 1)) * CX + cx]);
            m = fminf(m, ct[(cz * CY + (cy + 1)) * CX + cx]);
            m = fminf(m, ct[((cz - 1) * CY + cy) * CX + cx]);
            m = fminf(m, ct[((cz + 1) * CY + cy) * CX + cx]);
            v = m;
        }
        et[h] = v;
    }
    __syncthreads();

    int tx = tid % TX;   // 0..31
    int ty = tid / TX;   // 0..7
    #pragma unroll
    for (int oz = 0; oz < TZ; ++oz) {
        float m = -INFINITY;
        #pragma unroll
        for (int dz = 0; dz < 3; ++dz)
            #pragma unroll
            for (int dy = 0; dy < 3; ++dy)
                #pragma unroll
                for (int dx = 0; dx < 3; ++dx)
                    m = fmaxf(m, et[((oz + dz) * EY + (ty + dy)) * EX + (tx + dx)]);

        size_t gi = (((size_t)b * DIM + (z0 + oz)) * DIM + (y0 + ty)) * DIM + (x0 + tx);
        float c = ct[((oz + 2) * CY + (ty + 2)) * CX + (tx + 2)];  // cur, center
        float delta = fmaxf(c - m, 0.0f);
        float s;
        if (first) {
            s = delta;
        } else {
            float sk = skel[gi];
            s = sk + fmaxf(delta - sk * delta, 0.0f);
        }
        skel[gi] = s;
        if (write_e)
            e_out[gi] = et[((oz + 1) * EY + (ty + 1)) * EX + (tx + 1)];
    }
}

// ---------------------------------------------------------------------------
// Deterministic two-stage reduction (no float atomics -> bit-stable replays).
// partials[bid]              = sum(skel)
// partials[RED_BLOCKS + bid] = sum(skel * other), other = tgt or sigmoid(logit)
__global__ __launch_bounds__(256) void reduce_kernel(const float* __restrict__ skel,
                                                     const void* __restrict__ other,
                                                     int mode,
                                                     float* __restrict__ partials) {
    float s = 0.0f, sm = 0.0f;
    const int NV = NTOT / 4;
    for (int i = blockIdx.x * 256 + threadIdx.x; i < NV; i += RED_BLOCKS * 256) {
        float4 sk = ((const float4*)skel)[i];
        float4 o;
        if (mode == 0) {
            int4 t = ((const int4*)other)[i];
            o.x = (float)t.x; o.y = (float)t.y; o.z = (float)t.z; o.w = (float)t.w;
        } else {
            float4 l = ((const float4*)other)[i];
            o.x = sigmoidf_dev(l.x); o.y = sigmoidf_dev(l.y);
            o.z = sigmoidf_dev(l.z); o.w = sigmoidf_dev(l.w);
        }
        s += sk.x + sk.y + sk.z + sk.w;
        sm += sk.x * o.x + sk.y * o.y + sk.z * o.z + sk.w * o.w;
    }
    __shared__ float sh0[256];
    __shared__ float sh1[256];
    sh0[threadIdx.x] = s;
    sh1[threadIdx.x] = sm;
    __syncthreads();
    for (int w = 128; w > 0; w >>= 1) {
        if ((int)threadIdx.x < w) {
            sh0[threadIdx.x] += sh0[threadIdx.x + w];
            sh1[threadIdx.x] += sh1[threadIdx.x + w];
        }
        __syncthreads();
    }
    if (threadIdx.x == 0) {
        partials[blockIdx.x] = sh0[0];
        partials[RED_BLOCKS + blockIdx.x] = sh1[0];
    }
}

// segments: 0 = sum(skelP), 1 = sum(skelP*tgt), 2 = sum(skelT), 3 = sum(skelT*p)
__global__ __launch_bounds__(256) void final_kernel(const float* __restrict__ partials,
                                                    float* __restrict__ out) {
    __shared__ float sh[256];
    float sums[4];
    for (int seg = 0; seg < 4; ++seg) {
        float v = 0.0f;
        for (int i = threadIdx.x; i < RED_BLOCKS; i += 256)
            v += partials[seg * RED_BLOCKS + i];
        sh[threadIdx.x] = v;
        __syncthreads();
        for (int w = 128; w > 0; w >>= 1) {
            if ((int)threadIdx.x < w) sh[threadIdx.x] += sh[threadIdx.x + w];
            __syncthreads();
        }
        sums[seg] = sh[0];
        __syncthreads();
    }
    if (threadIdx.x == 0) {
        float tprec = (sums[1] + 1.0f) / (sums[0] + 1.0f);
        float tsens = (sums[3] + 1.0f) / (sums[2] + 1.0f);
        out[0] = 1.0f - 2.0f * (tprec * tsens) / (tprec + tsens);
    }
}

// ---------------------------------------------------------------------------
extern "C" void kernel_launch(void* const* d_in, const int* in_sizes, int n_in,
                              void* d_out, int out_size, void* d_ws, size_t ws_size,
                              hipStream_t stream) {
    (void)in_sizes; (void)n_in; (void)out_size; (void)ws_size;
    const float* logits = (const float*)d_in[0];
    const int* target = (const int*)d_in[1];
    float* out = (float*)d_out;

    float* ws = (float*)d_ws;
    float* A = ws;                    // current image
    float* B = ws + (size_t)NTOT;     // erode output / next cur
    float* S = ws + (size_t)2 * NTOT; // skeleton
    float* P = ws + (size_t)3 * NTOT; // 4*RED_BLOCKS partial sums

    const int iblk = (NTOT / 4) / 256;               // 8000
    dim3 tgrid(DIM / TX, DIM / TY, 2 * (DIM / TZ));  // (5,20,40)

    // ---- field P: p = sigmoid(logits) -------------------------------------
    init_field_kernel<<<iblk, 256, 0, stream>>>((const void*)logits, A, 0);
    {
        float* curv = A;
        float* er = B;
        for (int k = 0; k < 4; ++k) {
            fused_open_update_kernel<<<tgrid, 256, 0, stream>>>(
                curv, er, S, k == 0 ? 1 : 0, k < 3 ? 1 : 0);
            float* tmp = curv; curv = er; er = tmp;
        }
    }
    reduce_kernel<<<RED_BLOCKS, 256, 0, stream>>>(S, (const void*)target, 0, P);

    // ---- field T: tgt -----------------------------------------------------
    init_field_kernel<<<iblk, 256, 0, stream>>>((const void*)target, A, 1);
    {
        float* curv = A;
        float* er = B;
        for (int k = 0; k < 4; ++k) {
            fused_open_update_kernel<<<tgrid, 256, 0, stream>>>(
                curv, er, S, k == 0 ? 1 : 0, k < 3 ? 1 : 0);
            float* tmp = curv; curv = er; er = tmp;
        }
    }
    reduce_kernel<<<RED_BLOCKS, 256, 0, stream>>>(S, (const void*)logits, 1, P + 2 * RED_BLOCKS);

    final_kernel<<<1, 256, 0, stream>>>(P, out);
}